// LinOSSLayer_17978733101087
// MI455X (gfx1250) — compile-verified
//
#include <hip/hip_runtime.h>
#include <math.h>

// ---------------- problem dimensions ----------------
#define L_SEQ 16384
#define H_DIM 128
#define P_DIM 256
#define NCHUNK 256
#define CHUNK  64          // L_SEQ / NCHUNK

typedef __attribute__((ext_vector_type(2))) float v2f;
typedef __attribute__((ext_vector_type(8))) float v8f;

// ---------------- workspace layout (float offsets) ----------------
#define O_BRE   0
#define O_BIM   (O_BRE  + P_DIM*H_DIM)
#define O_CRE   (O_BIM  + P_DIM*H_DIM)
#define O_CIMN  (O_CRE  + H_DIM*P_DIM)
#define O_COEF  (O_CIMN + H_DIM*P_DIM)
#define O_CARRY (O_COEF + P_DIM*16)
#define O_BU    (O_CARRY + NCHUNK*P_DIM*2*2)
// total floats used: O_BU + 2*L_SEQ*P_DIM  (~35 MB)

// ---------------- setup: split B/C, build per-channel scan coefficients ----------------
__global__ void linoss_setup(const float* __restrict__ A_diag,
                             const float* __restrict__ B,
                             const float* __restrict__ C,
                             const float* __restrict__ steps,
                             float* __restrict__ ws) {
  int tid = blockIdx.x * blockDim.x + threadIdx.x;
  const int nBH = P_DIM * H_DIM;
  if (tid < nBH) {
    ws[O_BRE  + tid] =  B[2*tid + 0];
    ws[O_BIM  + tid] =  B[2*tid + 1];
    ws[O_CRE  + tid] =  C[2*tid + 0];
    ws[O_CIMN + tid] = -C[2*tid + 1];   // pre-negate: f32 WMMA has no A/B-neg, only CNeg
  }
  if (tid < P_DIM) {
    float A  = fmaxf(A_diag[tid], 0.0f);
    float dt = 1.0f / (1.0f + __expf(-steps[tid]));
    float schur = 1.0f / (1.0f + dt*dt*A);
    float M11 = 1.0f - dt*dt*A*schur;
    float M12 = -dt*A*schur;
    float M21 = dt*schur;
    float M22 = schur;
    // N = M^CHUNK (chunk-jump matrix for the carry scan)
    float N11=1.f, N12=0.f, N21=0.f, N22=1.f;
    #pragma unroll 1
    for (int i = 0; i < CHUNK; ++i) {
      float t11 = M11*N11 + M12*N21;
      float t12 = M11*N12 + M12*N22;
      float t21 = M21*N11 + M22*N21;
      float t22 = M21*N12 + M22*N22;
      N11=t11; N12=t12; N21=t21; N22=t22;
    }
    float* cf = ws + O_COEF + tid*16;
    cf[0]=M11; cf[1]=M12; cf[2]=M21; cf[3]=M22;
    cf[4]=M11*dt; cf[5]=M21*dt;      // F1 = c1*Bu, F2 = c2*Bu
    cf[6]=N11; cf[7]=N12; cf[8]=N21; cf[9]=N22;
  }
}

// helper: one f32 WMMA step
__device__ __forceinline__ v8f wmma4(v2f a, v2f b, v8f c) {
  return __builtin_amdgcn_wmma_f32_16x16x4_f32(false, a, false, b, (short)0, c, false, false);
}

// ---------------- GEMM1: Bu = X @ B^T (re & im) ----------------
// one wave -> 32x32 output tile per plane: 8 independent WMMA chains,
// 6 loads feed 8 WMMAs per K-step (A shared across n, B shared across m).
__global__ void linoss_gemm1(const float* __restrict__ X, float* __restrict__ ws) {
  const int wid  = threadIdx.x >> 5;
  const int lane = threadIdx.x & 31;
  const int m0 = blockIdx.x * 32;
  const int n0 = (blockIdx.y * 4 + wid) * 32;
  const float* Bre = ws + O_BRE;
  const float* Bim = ws + O_BIM;
  float* BuRe = ws + O_BU;
  float* BuIm = ws + O_BU + (size_t)L_SEQ * P_DIM;

  const int lr    = lane & 15;
  const int khalf = (lane >> 4) << 1;         // upper lane half holds K+2,K+3
  const float* xr0 = X + (size_t)(m0 + lr) * H_DIM + khalf;
  const float* xr1 = X + (size_t)(m0 + 16 + lr) * H_DIM + khalf;
  const float* br0p = Bre + (size_t)(n0 + lr) * H_DIM + khalf;
  const float* br1p = Bre + (size_t)(n0 + 16 + lr) * H_DIM + khalf;
  const float* bi0p = Bim + (size_t)(n0 + lr) * H_DIM + khalf;
  const float* bi1p = Bim + (size_t)(n0 + 16 + lr) * H_DIM + khalf;

  v8f aR00 = {}, aR01 = {}, aR10 = {}, aR11 = {};
  v8f aI00 = {}, aI01 = {}, aI10 = {}, aI11 = {};
  #pragma unroll 4
  for (int kk = 0; kk < H_DIM; kk += 4) {
    v2f a0  = *(const v2f*)(xr0 + kk);
    v2f a1  = *(const v2f*)(xr1 + kk);
    v2f br0 = *(const v2f*)(br0p + kk);
    v2f br1 = *(const v2f*)(br1p + kk);
    v2f bi0 = *(const v2f*)(bi0p + kk);
    v2f bi1 = *(const v2f*)(bi1p + kk);
    aR00 = wmma4(a0, br0, aR00);
    aR01 = wmma4(a0, br1, aR01);
    aR10 = wmma4(a1, br0, aR10);
    aR11 = wmma4(a1, br1, aR11);
    aI00 = wmma4(a0, bi0, aI00);
    aI01 = wmma4(a0, bi1, aI01);
    aI10 = wmma4(a1, bi0, aI10);
    aI11 = wmma4(a1, bi1, aI11);
  }
  const int rb = ((lane >> 4) << 3);          // D: VGPR r -> row r (+8 upper half)
  #pragma unroll
  for (int r = 0; r < 8; ++r) {
    size_t i00 = (size_t)(m0 + rb + r) * P_DIM + (n0 + lr);
    size_t i01 = i00 + 16;
    size_t i10 = i00 + (size_t)16 * P_DIM;
    size_t i11 = i10 + 16;
    BuRe[i00] = aR00[r]; BuRe[i01] = aR01[r];
    BuRe[i10] = aR10[r]; BuRe[i11] = aR11[r];
    BuIm[i00] = aI00[r]; BuIm[i01] = aI01[r];
    BuIm[i10] = aI10[r]; BuIm[i11] = aI11[r];
  }
}

// ---------------- scan phase 1: chunk-local affine offsets ----------------
__global__ void linoss_scan1(float* __restrict__ ws) {
  const int p     = threadIdx.x;          // 0..255 -> coalesced over Bu row
  const int part  = blockIdx.x & 1;
  const int chunk = blockIdx.x >> 1;
  const float* cf = ws + O_COEF + p*16;
  const float M11=cf[0], M12=cf[1], M21=cf[2], M22=cf[3], c1=cf[4], c2=cf[5];
  const float* Bu = ws + O_BU + (size_t)part * L_SEQ * P_DIM;
  float b1 = 0.f, b2 = 0.f;
  const int t0 = chunk * CHUNK;
  #pragma unroll 4
  for (int i = 0; i < CHUNK; ++i) {
    float u  = Bu[(size_t)(t0 + i) * P_DIM + p];
    float n1 = M11*b1 + M12*b2 + c1*u;
    float n2 = M21*b1 + M22*b2 + c2*u;
    b1 = n1; b2 = n2;
  }
  float* carry = ws + O_CARRY + (((size_t)chunk * P_DIM + p) * 2 + part) * 2;
  carry[0] = b1; carry[1] = b2;
}

// ---------------- scan phase 2: sequential carry composition (tiny) ----------------
__global__ void linoss_scan2(float* __restrict__ ws) {
  const int tid  = threadIdx.x;
  const int p    = tid & (P_DIM - 1);
  const int part = tid >> 8;
  const float* cf = ws + O_COEF + p*16;
  const float N11=cf[6], N12=cf[7], N21=cf[8], N22=cf[9];
  float s1 = 0.f, s2 = 0.f;
  #pragma unroll 1
  for (int c = 0; c < NCHUNK; ++c) {
    float* carry = ws + O_CARRY + (((size_t)c * P_DIM + p) * 2 + part) * 2;
    float b1 = carry[0], b2 = carry[1];
    carry[0] = s1; carry[1] = s2;               // exclusive prefix state
    float t1 = N11*s1 + N12*s2 + b1;
    float t2 = N21*s1 + N22*s2 + b2;
    s1 = t1; s2 = t2;
  }
}

// ---------------- scan phase 3: replay chunks, write x2 in place over Bu ----------------
__global__ void linoss_scan3(float* __restrict__ ws) {
  const int p     = threadIdx.x;
  const int part  = blockIdx.x & 1;
  const int chunk = blockIdx.x >> 1;
  const float* cf = ws + O_COEF + p*16;
  const float M11=cf[0], M12=cf[1], M21=cf[2], M22=cf[3], c1=cf[4], c2=cf[5];
  const float* carry = ws + O_CARRY + (((size_t)chunk * P_DIM + p) * 2 + part) * 2;
  float x1 = carry[0], x2 = carry[1];
  float* Bu = ws + O_BU + (size_t)part * L_SEQ * P_DIM;
  const int t0 = chunk * CHUNK;
  #pragma unroll 4
  for (int i = 0; i < CHUNK; ++i) {
    size_t idx = (size_t)(t0 + i) * P_DIM + p;
    float u  = Bu[idx];
    float n1 = M11*x1 + M12*x2 + c1*u;
    float n2 = M21*x1 + M22*x2 + c2*u;
    x1 = n1; x2 = n2;
    Bu[idx] = x2;                               // x2 plane replaces Bu plane
  }
}

// ---------------- GEMM2: ys = x2re @ Cre^T + x2im @ (-Cim)^T + X*D ----------------
// 32x32 tile per wave, 4 accumulators, re+im folded -> 8 WMMAs per K-step.
__global__ void linoss_gemm2(const float* __restrict__ X,
                             const float* __restrict__ Dvec,
                             const float* __restrict__ ws,
                             float* __restrict__ out) {
  const int wid  = threadIdx.x >> 5;
  const int lane = threadIdx.x & 31;
  const int m0 = blockIdx.x * 32;
  const int n0 = wid * 32;                     // H/32 = 4 n-tiles, one per wave
  const float* Cre  = ws + O_CRE;
  const float* CimN = ws + O_CIMN;
  const float* X2Re = ws + O_BU;
  const float* X2Im = ws + O_BU + (size_t)L_SEQ * P_DIM;

  const int lr    = lane & 15;
  const int khalf = (lane >> 4) << 1;
  const float* ar0p = X2Re + (size_t)(m0 + lr) * P_DIM + khalf;
  const float* ar1p = X2Re + (size_t)(m0 + 16 + lr) * P_DIM + khalf;
  const float* ai0p = X2Im + (size_t)(m0 + lr) * P_DIM + khalf;
  const float* ai1p = X2Im + (size_t)(m0 + 16 + lr) * P_DIM + khalf;
  const float* br0p = Cre  + (size_t)(n0 + lr) * P_DIM + khalf;
  const float* br1p = Cre  + (size_t)(n0 + 16 + lr) * P_DIM + khalf;
  const float* bi0p = CimN + (size_t)(n0 + lr) * P_DIM + khalf;
  const float* bi1p = CimN + (size_t)(n0 + 16 + lr) * P_DIM + khalf;

  v8f a00 = {}, a01 = {}, a10 = {}, a11 = {};
  #pragma unroll 4
  for (int kk = 0; kk < P_DIM; kk += 4) {
    v2f ar0 = *(const v2f*)(ar0p + kk);
    v2f ar1 = *(const v2f*)(ar1p + kk);
    v2f ai0 = *(const v2f*)(ai0p + kk);
    v2f ai1 = *(const v2f*)(ai1p + kk);
    v2f br0 = *(const v2f*)(br0p + kk);
    v2f br1 = *(const v2f*)(br1p + kk);
    v2f bi0 = *(const v2f*)(bi0p + kk);
    v2f bi1 = *(const v2f*)(bi1p + kk);
    a00 = wmma4(ar0, br0, a00);
    a01 = wmma4(ar0, br1, a01);
    a10 = wmma4(ar1, br0, a10);
    a11 = wmma4(ar1, br1, a11);
    a00 = wmma4(ai0, bi0, a00);
    a01 = wmma4(ai0, bi1, a01);
    a10 = wmma4(ai1, bi0, a10);
    a11 = wmma4(ai1, bi1, a11);
  }
  const int rb = ((lane >> 4) << 3);
  #pragma unroll
  for (int r = 0; r < 8; ++r) {
    int row0 = m0 + rb + r;
    int row1 = row0 + 16;
    int c0 = n0 + lr, c1 = n0 + 16 + lr;
    size_t i00 = (size_t)row0 * H_DIM + c0;
    size_t i01 = (size_t)row0 * H_DIM + c1;
    size_t i10 = (size_t)row1 * H_DIM + c0;
    size_t i11 = (size_t)row1 * H_DIM + c1;
    out[i00] = a00[r] + X[i00] * Dvec[c0];
    out[i01] = a01[r] + X[i01] * Dvec[c1];
    out[i10] = a10[r] + X[i10] * Dvec[c0];
    out[i11] = a11[r] + X[i11] * Dvec[c1];
  }
}

// ---------------- launcher ----------------
extern "C" void kernel_launch(void* const* d_in, const int* in_sizes, int n_in,
                              void* d_out, int out_size, void* d_ws, size_t ws_size,
                              hipStream_t stream) {
  (void)in_sizes; (void)n_in; (void)out_size; (void)ws_size;
  const float* X      = (const float*)d_in[0];   // (L,H)
  const float* A_diag = (const float*)d_in[1];   // (P,)
  const float* B      = (const float*)d_in[2];   // (P,H,2)
  const float* C      = (const float*)d_in[3];   // (H,P,2)
  const float* Dvec   = (const float*)d_in[4];   // (H,)
  const float* steps  = (const float*)d_in[5];   // (P,)
  float* out = (float*)d_out;
  float* ws  = (float*)d_ws;

  // 1) split B/C, build scan coefficients (incl. M^CHUNK)
  linoss_setup<<<(P_DIM * H_DIM + 255) / 256, 256, 0, stream>>>(A_diag, B, C, steps, ws);

  // 2) Bu = X @ B^T (re/im), f32 WMMA, 32x32 tile/wave. Grid: (L/32) x (P/(32*4)).
  dim3 g1(L_SEQ / 32, P_DIM / 128);
  linoss_gemm1<<<g1, 128, 0, stream>>>(X, ws);

  // 3) chunked affine scan over L (512 planes x 256 chunks)
  linoss_scan1<<<NCHUNK * 2, P_DIM, 0, stream>>>(ws);
  linoss_scan2<<<1, P_DIM * 2, 0, stream>>>(ws);
  linoss_scan3<<<NCHUNK * 2, P_DIM, 0, stream>>>(ws);

  // 4) ys = Re(x2 @ C^T) + X*D, f32 WMMA, 32x32 tile/wave. Grid: (L/32) x 1.
  dim3 g2(L_SEQ / 32, 1);
  linoss_gemm2<<<g2, 128, 0, stream>>>(X, Dvec, ws, out);
}